// YOLOActivation_38414187495678
// MI455X (gfx1250) — compile-verified
//
#include <hip/hip_runtime.h>

#define SEG            85     // 80 classes + 5 box terms
#define NCELLS         49
#define SEGS_PER_BLOCK 8
#define THREADS        256    // 8 wave32s
#define TILE_FLOATS    (SEGS_PER_BLOCK * SEG)   // 680 floats = 2720 B, 16B-aligned per block
#define TILE_VEC4      (TILE_FLOATS / 4)        // 170 x b128 transfers

typedef float  vfloat4 __attribute__((ext_vector_type(4)));
// exact pointee type the async-LDS builtins expect (from the round-1 diagnostic):
typedef int    v4i_t   __attribute__((__vector_size__(4 * sizeof(int))));
typedef __attribute__((address_space(1))) v4i_t* global_v4i_ptr;
typedef __attribute__((address_space(3))) v4i_t* lds_v4i_ptr;

// CDNA5 async global->LDS DMA path (gfx1250). Guarded so the kernel still
// compiles if the toolchain lacks the builtins (falls back to b128 NT load
// + ds_store_b128, same data movement, synchronous).
#ifdef __HIP_DEVICE_COMPILE__
#  if __has_builtin(__builtin_amdgcn_global_load_async_to_lds_b128) && \
      __has_builtin(__builtin_amdgcn_s_wait_asynccnt)
#    define USE_ASYNC_LDS 1
#  endif
#endif
#ifndef USE_ASYNC_LDS
#  define USE_ASYNC_LDS 0
#endif

__global__ __launch_bounds__(THREADS)
void yolo_act_gfx1250_kernel(const float* __restrict__ in,
                             float* __restrict__ out,
                             int total_segs) {
  __shared__ float tile[TILE_FLOATS];

  const int tid  = threadIdx.x;
  const int lane = tid & 31;
  const int wave = tid >> 5;

  const long long blk_seg0    = (long long)blockIdx.x * SEGS_PER_BLOCK;
  const long long gbase       = blk_seg0 * SEG;                 // tile start (elements)
  const long long total_elems = (long long)total_segs * SEG;

  // ---- prefetch a future tile into L2 (global_prefetch_b8), ~8 blocks ahead ----
  {
    long long pf = gbase + (long long)8 * TILE_FLOATS + (long long)tid * 64; // 256B/lane
    if (tid < 11 && pf < total_elems)
      __builtin_prefetch(in + pf, 0, 1);
  }

  // ---- cooperative tile load: global -> LDS, 170 x 16B ----
  if (tid < TILE_VEC4) {
    const long long ge = gbase + (long long)tid * 4;
    if (ge + 4 <= total_elems) {
#if USE_ASYNC_LDS
      __builtin_amdgcn_global_load_async_to_lds_b128(
          (global_v4i_ptr)(in + ge),
          (lds_v4i_ptr)(&tile[tid * 4]),
          /*offset=*/0, /*cpol=*/0);
#else
      *(vfloat4*)&tile[tid * 4] =
          __builtin_nontemporal_load((const vfloat4*)(in + ge));
#endif
    } else {
      // generic tail guard (not hit for the reference sizes: 802816 % 8 == 0)
      for (int k = 0; k < 4; ++k) {
        long long e = ge + k;
        tile[tid * 4 + k] = (e < total_elems) ? in[e] : 0.0f;
      }
    }
  }
#if USE_ASYNC_LDS
  __builtin_amdgcn_s_wait_asynccnt(0);   // this wave's DMA transfers landed in LDS
#endif
  __syncthreads();                       // all waves' tile data visible

  // ---- one wave per 85-float segment ----
  const long long seg = blk_seg0 + wave;
  if (seg >= total_segs) return;

  const int cell = (int)(seg % NCELLS);
  const int off  = wave * SEG;

  // lane l owns elements l, l+32, l+64 (bank-conflict-free: (21w+l) mod 64)
  float v0 = tile[off + lane];                              // idx  0..31
  float v1 = tile[off + 32 + lane];                         // idx 32..63
  float v2 = (lane < 21) ? tile[off + 64 + lane] : 0.0f;    // idx 64..84

  float r0 = v0, r1 = v1, r2 = v2;                          // cell 48: plain copy

  if (cell != NCELLS - 1) {
    const float NEG = -__builtin_inff();
    // classes are idx 5..84: v1 always class; v0 iff lane>=5; v2 iff lane<21
    float m = v1;
    m = fmaxf(m, (lane >= 5) ? v0 : NEG);
    m = fmaxf(m, (lane < 21) ? v2 : NEG);
#pragma unroll
    for (int o = 16; o > 0; o >>= 1) m = fmaxf(m, __shfl_xor(m, o, 32));

    float e0 = (lane >= 5) ? __expf(v0 - m) : 0.0f;
    float e1 = __expf(v1 - m);
    float e2 = (lane < 21) ? __expf(v2 - m) : 0.0f;
    float s  = e0 + e1 + e2;
#pragma unroll
    for (int o = 16; o > 0; o >>= 1) s += __shfl_xor(s, o, 32);

    const float inv = __builtin_amdgcn_rcpf(s);
    r1 = e1 * inv;
    r2 = e2 * inv;
    if (lane >= 5) {
      r0 = e0 * inv;                                        // class prob
    } else if (lane == 2 || lane == 3) {
      r0 = v0;                                              // wh: copy
    } else {
      r0 = __builtin_amdgcn_rcpf(1.0f + __expf(-v0));       // xy/conf: sigmoid
    }
  }

  // ---- streaming NT stores (no reuse -> keep L2 for the input stream) ----
  float* q = out + seg * SEG;
  __builtin_nontemporal_store(r0, q + lane);
  __builtin_nontemporal_store(r1, q + 32 + lane);
  if (lane < 21) __builtin_nontemporal_store(r2, q + 64 + lane);
}

extern "C" void kernel_launch(void* const* d_in, const int* in_sizes, int n_in,
                              void* d_out, int out_size, void* d_ws, size_t ws_size,
                              hipStream_t stream) {
  const float* in  = (const float*)d_in[0];
  float*       out = (float*)d_out;

  const long long total_elems = (long long)in_sizes[0];
  const int total_segs = (int)(total_elems / SEG);           // 16384*49 = 802816
  const int blocks = (total_segs + SEGS_PER_BLOCK - 1) / SEGS_PER_BLOCK;

  yolo_act_gfx1250_kernel<<<blocks, THREADS, 0, stream>>>(in, out, total_segs);
}